// MacTensorUnit_12893491823233
// MI455X (gfx1250) — compile-verified
//
#include <hip/hip_runtime.h>
#include <math.h>

typedef __attribute__((ext_vector_type(2))) float v2f;
typedef __attribute__((ext_vector_type(8))) float v8f;

#define BATCH      256
#define IN_C       256
#define IN_S       196
#define C_DIM      512
#define S_DIM      196
#define OUT_S      49
#define F_PER_B    (C_DIM * OUT_S)        /* 25088, divisible by 16 */
#define TILES_PER_B (F_PER_B / 16)        /* 1568 */
#define TOTAL_TASKS (BATCH * TILES_PER_B) /* 401408 */

// Piecewise-linear 5-entry LUT lookup, faithful to reference clamping.
// Branch-free: bgn in [0,3]; selects lower as v_cndmask chains.
__device__ __forceinline__ float lut5(float m0, float m1, float m2, float m3,
                                      float m4, float sig) {
  float idx = sig * 5.0f;
  int bgn = (int)floorf(idx);
  bgn = bgn < 0 ? 0 : bgn;
  bgn = bgn > 3 ? bgn - 1 : bgn;   // applied twice, like reference
  bgn = bgn > 3 ? bgn - 1 : bgn;
  float pos = idx - (float)bgn;
  float lo = (bgn == 0) ? m0 : ((bgn == 1) ? m1 : ((bgn == 2) ? m2 : m3));
  float hi = (bgn == 0) ? m1 : ((bgn == 1) ? m2 : ((bgn == 2) ? m3 : m4));
  return (1.0f - pos) * lo + pos * hi;
}

// Full per-element pipeline: 3 MAC steps + attention gate.
__device__ __forceinline__ float transform(float x,
    float a0, float a1, float a2, float a3, float a4,
    float v0, float v1, float v2, float v3, float v4,
    float ow, float ob) {
  const float S2 = 0.33333f * 0.33333f;  // step() * nonlinear() double scale
#pragma unroll
  for (int i = 0; i < 3; ++i) {
    float sig_a = 0.5f * (1.0f + erff(x * 0.70710678118654752f));  // ngd
    float ang   = lut5(a0, a1, a2, a3, a4, sig_a);
    float sig_v = 0.5f * (1.0f + erff(x));                         // nerf
    float vel   = lut5(v0, v1, v2, v3, v4, sig_v);
    float ca = __cosf(ang);   // v_cos_f32
    float sa = __sinf(ang);   // v_sin_f32
    x = x + (vel * ca + x * vel * sa) * S2;
  }
  // alg1 gate: g/(1+|g|) via v_rcp_f32 instead of IEEE divide sequence.
  float g   = x * ow + ob;
  float r   = __builtin_amdgcn_rcpf(1.0f + fabsf(g));  // v_rcp_f32
  float att = 0.5f * (g * r) + 0.5f;
  return x * att;
}

__global__ void __launch_bounds__(256)
MacTensorUnit_12893491823233_kernel(
    const float* __restrict__ data,
    const float* __restrict__ in_w,  const float* __restrict__ in_b,
    const float* __restrict__ out_w, const float* __restrict__ out_b,
    const float* __restrict__ ang,   const float* __restrict__ vel,
    float* __restrict__ out) {
  const int lane        = threadIdx.x & 31;
  const int waveInBlk   = threadIdx.x >> 5;
  const int wavesPerBlk = blockDim.x >> 5;
  const int waveId      = blockIdx.x * wavesPerBlk + waveInBlk;
  const int nWaves      = gridDim.x * wavesPerBlk;

  // 5-entry LUT tables held in registers (uniform loads).
  const float a0 = ang[0], a1 = ang[1], a2 = ang[2], a3 = ang[3], a4 = ang[4];
  const float v0 = vel[0], v1 = vel[1], v2 = vel[2], v3 = vel[3], v4 = vel[4];

  // A = 16x4 matrix of ones -> D[m,n] = sum_k B[k,n] (column sums).
  v2f A;  A.x = 1.0f;  A.y = 1.0f;

  // B-matrix slot for this lane: column n = lane&15, K rows {k0, k0+1}.
  const int n  = lane & 15;
  const int k0 = (lane >> 4) << 1;  // 0 for lanes 0-15, 2 for lanes 16-31

  for (int task = waveId; task < TOTAL_TASKS; task += nWaves) {
    const int b     = task / TILES_PER_B;
    const int fbase = (task - b * TILES_PER_B) << 4;  // 16 outputs per tile
    const int f     = fbase + n;                      // flat (c,so)
    const int c     = f / OUT_S;
    const int so    = f - c * OUT_S;

    const size_t drow = ((size_t)b * IN_C + (size_t)(c >> 1)) * (size_t)IN_S;
    const int    cs   = c * S_DIM;
    const int    s0   = k0 * OUT_S + so;   // spatial index for sf = k0
    const int    s1   = s0 + OUT_S;        // spatial index for sf = k0+1

    // Expansion (mad), then the heavy elementwise pipeline, per K slot.
    float x0 = data[drow + s0] * in_w[cs + s0] + in_b[cs + s0];
    float x1 = data[drow + s1] * in_w[cs + s1] + in_b[cs + s1];
    float y0 = transform(x0, a0, a1, a2, a3, a4, v0, v1, v2, v3, v4,
                         out_w[cs + s0], out_b[cs + s0]);
    float y1 = transform(x1, a0, a1, a2, a3, a4, v0, v1, v2, v3, v4,
                         out_w[cs + s1], out_b[cs + s1]);

    // Cross-lane sf-reduction on the matrix unit:
    // D = ones(16x4) x B(4x16) -> every row of D holds the 16 column sums.
    v2f Bm;  Bm.x = y0;  Bm.y = y1;
    v8f Cm = {};
    v8f D = __builtin_amdgcn_wmma_f32_16x16x4_f32(
        /*neg_a=*/false, A, /*neg_b=*/false, Bm,
        /*c_mod=*/(short)0, Cm, /*reuse_a=*/false, /*reuse_b=*/false);

    // D VGPR0, lanes 0-15 = row M=0, columns N=0..15.
    if (lane < 16) {
      out[(size_t)b * F_PER_B + (size_t)(fbase + lane)] = D[0];
    }
  }
}

extern "C" void kernel_launch(void* const* d_in, const int* in_sizes, int n_in,
                              void* d_out, int out_size, void* d_ws,
                              size_t ws_size, hipStream_t stream) {
  const float* data  = (const float*)d_in[0];
  const float* in_w  = (const float*)d_in[1];
  const float* in_b  = (const float*)d_in[2];
  const float* out_w = (const float*)d_in[3];
  const float* out_b = (const float*)d_in[4];
  const float* ang   = (const float*)d_in[5];
  const float* vel   = (const float*)d_in[6];
  float* out = (float*)d_out;

  // 12544 blocks x 8 waves = 100352 waves; 401408 tasks -> 4 tasks/wave.
  dim3 grid(12544), block(256);
  hipLaunchKernelGGL(MacTensorUnit_12893491823233_kernel, grid, block, 0,
                     stream, data, in_w, in_b, out_w, out_b, ang, vel, out);
}